// GateRecurrent2dnoind_60954175865171
// MI455X (gfx1250) — compile-verified
//
#include <hip/hip_runtime.h>
#include <stdint.h>

// Problem geometry (from reference): [B=16, C=32, H=128, W=128], float32.
#define HGT     128
#define WID     128
#define NPLANES (16 * 32)          // independent (batch, channel) planes
#define WT      4                  // width-tile: sequential steps per LDS tile
#define NT      (WID / WT)         // 32 tiles per plane
#define NIN     5                  // X, B, G1, G2, G3
#define TILEF   (HGT * WT)         // floats per input tile in LDS (512)
#define LDS_BYTES (2 * NIN * TILEF * 4)   // double-buffered: 20480 B / block

// ---- CDNA5 async global->LDS (VGLOBAL encoding, tracked by ASYNCcnt) ----
__device__ __forceinline__ void async_ld_b128(uint32_t lds_off, uint64_t gaddr) {
  // per-lane: copies 16 bytes from global address gaddr to LDS address lds_off
  asm volatile("global_load_async_to_lds_b128 %0, %1, off"
               :: "v"(lds_off), "v"(gaddr)
               : "memory");
}
__device__ __forceinline__ void wait_async_le20() {
  asm volatile("s_wait_asynccnt 20" ::: "memory");
}
__device__ __forceinline__ void wait_async_le0() {
  asm volatile("s_wait_asynccnt 0" ::: "memory");
}

extern __shared__ float smem[];    // 2 buffers x 5 inputs x [128 h][4 w]

__global__ __launch_bounds__(32)
void spn_scan_kernel(const float* __restrict__ X,  const float* __restrict__ B,
                     const float* __restrict__ G1, const float* __restrict__ G2,
                     const float* __restrict__ G3, float* __restrict__ Out)
{
  const uint32_t lane  = threadIdx.x;          // one wave32 per block/plane
  const uint32_t plane = blockIdx.x;           // 0..511
  const uint64_t planeBytes = (uint64_t)plane * (HGT * WID) * 4u;

  const float* ins[NIN] = {X, B, G1, G2, G3};

  // Low 32 bits of a flat LDS pointer == workgroup-relative LDS byte address.
  const uint32_t ldsBase = (uint32_t)(uintptr_t)smem;

  // Issue one tile (t) of all 5 inputs into LDS buffer `buf`.
  // Per input: 128 rows x 16 B = 4 wave-instructions (32 rows each) -> 20/tile.
  auto issue = [&](int t, int buf) {
#pragma unroll
    for (int i = 0; i < NIN; ++i) {
      uint64_t g = (uint64_t)(uintptr_t)ins[i] + planeBytes
                 + (uint64_t)lane * (WID * 4u) + (uint64_t)t * (WT * 4u);
      uint32_t lo = ldsBase + ((uint32_t)(buf * NIN + i) * TILEF + lane * WT) * 4u;
#pragma unroll
      for (int j = 0; j < 4; ++j) {            // rows 32j .. 32j+31
        async_ld_b128(lo, g);
        g  += 32u * WID * 4u;                  // 32 rows forward in global
        lo += 32u * WT  * 4u;                  // 32 rows forward in LDS
      }
    }
  };

  float hprev[4];                              // H[., w-1] for h = lane + 32k
#pragma unroll
  for (int k = 0; k < 4; ++k) hprev[k] = 0.0f;

  issue(0, 0);

  for (int t = 0; t < NT; ++t) {
    const int buf = t & 1;
    if (t + 1 < NT) { issue(t + 1, buf ^ 1); wait_async_le20(); }  // tile t landed
    else            { wait_async_le0(); }

    const uint32_t bufBase = (uint32_t)(buf * NIN) * TILEF;
    float oacc[4][WT];                         // output accum per k, per w-in-tile

#pragma unroll
    for (int s2 = 0; s2 < WT / 2; ++s2) {      // read two w-steps per DS b64 load
      float2 x2[4], b2[4], q1[4], q2[4], q3[4];
#pragma unroll
      for (int k = 0; k < 4; ++k) {
        const uint32_t row = (lane + 32u * k) * WT + 2u * s2;
        x2[k] = *(const float2*)&smem[bufBase + 0 * TILEF + row];
        b2[k] = *(const float2*)&smem[bufBase + 1 * TILEF + row];
        q1[k] = *(const float2*)&smem[bufBase + 2 * TILEF + row];
        q2[k] = *(const float2*)&smem[bufBase + 3 * TILEF + row];
        q3[k] = *(const float2*)&smem[bufBase + 4 * TILEF + row];
      }
#pragma unroll
      for (int half = 0; half < 2; ++half) {
        const int s = 2 * s2 + half;
        // Rotate previous column across the wave for h-1 / h+1 neighbors.
        float rotu[4], rotd[4];
#pragma unroll
        for (int k = 0; k < 4; ++k)
          rotu[k] = __shfl(hprev[k], (int)((lane + 31u) & 31u), 32);
#pragma unroll
        for (int k = 0; k < 4; ++k)
          rotd[k] = __shfl(hprev[k], (int)((lane + 1u) & 31u), 32);

        float up[4], dn[4];
        up[0] = (lane == 0u) ? 0.0f : rotu[0];
#pragma unroll
        for (int k = 1; k < 4; ++k) up[k] = (lane == 0u)  ? rotu[k - 1] : rotu[k];
#pragma unroll
        for (int k = 0; k < 3; ++k) dn[k] = (lane == 31u) ? rotd[k + 1] : rotd[k];
        dn[3] = (lane == 31u) ? 0.0f : rotd[3];

#pragma unroll
        for (int k = 0; k < 4; ++k) {
          const float xv = half ? x2[k].y : x2[k].x;
          const float bv = half ? b2[k].y : b2[k].x;
          const float a1 = half ? q1[k].y : q1[k].x;
          const float a2 = half ? q2[k].y : q2[k].x;
          const float a3 = half ? q3[k].y : q3[k].x;
          float h = bv * xv;
          h = fmaf(a1, up[k],    h);
          h = fmaf(a2, hprev[k], h);
          h = fmaf(a3, dn[k],    h);
          hprev[k]   = h;
          oacc[k][s] = h;
        }
      }
    }

    // Coalesced 16 B stores: one float4 per owned height.
#pragma unroll
    for (int k = 0; k < 4; ++k) {
      float* dst = Out + (size_t)plane * (HGT * WID)
                       + (size_t)(lane + 32u * k) * WID + (size_t)t * WT;
      *(float4*)dst = *(float4*)&oacc[k][0];
    }
  }
}

extern "C" void kernel_launch(void* const* d_in, const int* in_sizes, int n_in,
                              void* d_out, int out_size, void* d_ws, size_t ws_size,
                              hipStream_t stream) {
  const float* X  = (const float*)d_in[0];
  const float* B  = (const float*)d_in[1];
  const float* G1 = (const float*)d_in[2];
  const float* G2 = (const float*)d_in[3];
  const float* G3 = (const float*)d_in[4];
  float* Out = (float*)d_out;
  hipLaunchKernelGGL(spn_scan_kernel, dim3(NPLANES), dim3(32), LDS_BYTES, stream,
                     X, B, G1, G2, G3, Out);
}